// DeepFace_41102837022854
// MI455X (gfx1250) — compile-verified
//
#include <hip/hip_runtime.h>

typedef __attribute__((ext_vector_type(2))) float v2f;
typedef __attribute__((ext_vector_type(8))) float v8f;

static __device__ __forceinline__ v8f wmma4(v2f a, v2f b, v8f c) {
  // D = A(16x4 f32) * B(4x16 f32) + C(16x16 f32)  -> v_wmma_f32_16x16x4_f32
  return __builtin_amdgcn_wmma_f32_16x16x4_f32(false, a, false, b, (short)0, c, false, false);
}

// Pad K so 16 lanes with stride KPAD hit 64 distinct LDS banks on ds_load_b64:
// KPAD ≡ 4 (mod 8)  ->  gcd(KPAD,64)=4, banks 4*(m mod 16)+{0..3} all distinct.
template <int K> struct KPadT { static constexpr int value = K + ((4 - (K & 7)) & 7); };

// =====================================================================
// Generic im2col-tile WMMA conv (+bias+relu), K chunked by CCH channels.
//   M = B*HO*WO rows (16-row tiles, MT tiles/block), N = CO (16-col tiles).
//   Stage: LDS As[mt][16][KPAD] per chunk, row-wise KS-contiguous copies.
//   Main:  ds_load_b64 (A) + global_load_b64 (B, L2-resident) + wmma.
// =====================================================================
template <int KS, int CIN, int CCH, int HI, int WI, int HO, int WO, int CO, int MT>
__global__ __launch_bounds__((CO / 16) * MT * 32) void k_conv(
    const float* __restrict__ x, const float* __restrict__ w,
    const float* __restrict__ bias, float* __restrict__ y) {
  constexpr int NT     = CO / 16;
  constexpr int KFULL  = CIN * KS * KS;
  constexpr int KCH    = CCH * KS * KS;
  constexpr int KPAD   = KPadT<KCH>::value;
  constexpr int NTHR   = NT * MT * 32;
  extern __shared__ float As[];  // MT * 16 * KPAD floats

  const int lane  = threadIdx.x & 31;
  const int lm    = lane & 15;   // A row (within tile) / B col (out channel)
  const int khalf = lane >> 4;
  const int wb    = threadIdx.x >> 5;
  const int mt    = wb / NT;           // which M tile this wave works on
  const int n0    = (wb % NT) * 16;    // N tile base

  v8f acc = {};
  for (int ch = 0; ch < CIN / CCH; ++ch) {
    __syncthreads();  // protect previous chunk's LDS reads
    // ---- stage chunk: rows (mt_, m, c, i), each KS contiguous floats ----
    for (int r = threadIdx.x; r < MT * 16 * CCH * KS; r += NTHR) {
      const int i  = r % KS;
      int t        = r / KS;
      const int c  = t % CCH;  t /= CCH;
      const int m  = t & 15;
      const int mt_ = t >> 4;
      const int mg = (blockIdx.x * MT + mt_) * 16 + m;  // global output row
      const int wo = mg % WO;
      const int t2 = mg / WO;
      const int ho = t2 % HO;
      const int b  = t2 / HO;
      const float* src =
          x + (((size_t)b * CIN + (ch * CCH + c)) * HI + (ho + i)) * WI + wo;
      float* dst = As + (size_t)(mt_ * 16 + m) * KPAD + (c * KS + i) * KS;
#pragma unroll
      for (int j = 0; j < KS; ++j) dst[j] = src[j];
    }
    __syncthreads();
    // ---- wmma over chunk ----
    const float* arow = As + (size_t)(mt * 16 + lm) * KPAD;
    const float* brow = w + (size_t)(n0 + lm) * KFULL + ch * KCH;  // OIHW row
    for (int kk = 0; kk < KCH; kk += 4) {
      const int k0 = kk + 2 * khalf;
      v2f a = *reinterpret_cast<const v2f*>(arow + k0);
      v2f b = *reinterpret_cast<const v2f*>(brow + k0);
      acc = wmma4(a, b, acc);
    }
  }
  // ---- store: C/D layout M = r + 8*khalf, N = lane&15 ----
  const float bv = bias[n0 + lm];
  const int mg_base = (blockIdx.x * MT + mt) * 16;
#pragma unroll
  for (int r = 0; r < 8; ++r) {
    const int mg = mg_base + r + 8 * khalf;
    const int wo = mg % WO;
    const int t2 = mg / WO;
    const int ho = t2 % HO;
    const int b  = t2 / HO;
    y[(((size_t)b * CO + (n0 + lm)) * HO + ho) * WO + wo] = fmaxf(acc[r] + bv, 0.0f);
  }
}

// ---------------- maxpool 3x3 stride 2 ----------------
// x[16][32][102][102] -> y[16][32][50][50]
__global__ __launch_bounds__(256) void k_pool(const float* __restrict__ x,
                                              float* __restrict__ y) {
  int idx = blockIdx.x * 256 + threadIdx.x;
  const int WO = 50, HO = 50, C = 32, HI = 102;
  int wo = idx % WO; int t = idx / WO;
  int ho = t % HO;   t /= HO;
  int c = t % C;     int b = t / C;
  const float* xr = x + ((b * C + c) * HI + 2 * ho) * HI + 2 * wo;
  float m = -3.402823466e38f;
#pragma unroll
  for (int i = 0; i < 3; ++i)
#pragma unroll
    for (int j = 0; j < 3; ++j) m = fmaxf(m, xr[i * HI + j]);
  y[idx] = m;
}

// =====================================================================
// Locally-connected layer via f32 WMMA, LDS-staged activation patch.
// One block per spatial position; NT = COUT/16 waves, one N-tile each.
// Weights (the HBM-dominant stream) are loaded non-temporally so the
// L2-resident activations aren't evicted.
// =====================================================================
template <int KS, int CIN, int WIN, int HOUT, int COUT>
__global__ __launch_bounds__((COUT / 16) * 32) void k_local(
    const float* __restrict__ xin, const float* __restrict__ wgt,
    const float* __restrict__ bias, float* __restrict__ out) {
  constexpr int K    = CIN * KS * KS;
  constexpr int KPAD = KPadT<K>::value;
  constexpr int NT   = COUT / 16;
  extern __shared__ float As[];  // 16 * KPAD floats

  const int pos = blockIdx.x;          // ph*HOUT + pw (square output)
  const int ph  = pos / HOUT;
  const int pw  = pos % HOUT;

  // ---- stage patch: rows (m, c, i), each KS contiguous floats ----
  for (int r = threadIdx.x; r < 16 * CIN * KS; r += NT * 32) {
    const int i = r % KS;
    int t       = r / KS;
    const int c = t % CIN;
    const int m = t / CIN;
    const float* src = xin + (((size_t)m * CIN + c) * WIN + (ph + i)) * WIN + pw;
    float* dst = As + (size_t)m * KPAD + (c * KS + i) * KS;
#pragma unroll
    for (int j = 0; j < KS; ++j) dst[j] = src[j];
  }
  __syncthreads();

  const int lane  = threadIdx.x & 31;
  const int lm    = lane & 15;
  const int khalf = lane >> 4;
  const int n0    = (threadIdx.x >> 5) * 16;

  const float* arow = As + (size_t)lm * KPAD;
  const float* brow = wgt + ((size_t)pos * COUT + (n0 + lm)) * (size_t)K;

  v8f acc = {};
  for (int kk = 0; kk < K; kk += 4) {
    const int k0 = kk + 2 * khalf;
    v2f a = *reinterpret_cast<const v2f*>(arow + k0);               // ds_load_b64
    v2f b = __builtin_nontemporal_load(
        reinterpret_cast<const v2f*>(brow + k0));                    // NT stream
    acc = wmma4(a, b, acc);
  }

  const float bv = bias[(size_t)pos * COUT + n0 + lm];
#pragma unroll
  for (int r = 0; r < 8; ++r) {
    const int bat = r + 8 * khalf;
    out[(((size_t)bat * COUT + (n0 + lm)) * HOUT + ph) * HOUT + pw] = acc[r] + bv;
  }
}

// ---------------- FC: split-K WMMA GEMM, atomic f32 accumulate ----------------
// out[16 x 1024] += X[16 x K] * W^T[K x 1024], K = 256*22*22 = 123904.
__global__ __launch_bounds__(128) void k_fc(const float* __restrict__ xin,
                                            const float* __restrict__ wgt,
                                            float* __restrict__ accum) {
  constexpr int K      = 256 * 22 * 22;   // 123904
  constexpr int NSPLIT = 64;
  constexpr int CHUNK  = K / NSPLIT;      // 1936, multiple of 4
  const int lane  = threadIdx.x & 31;
  const int wave  = blockIdx.x * 4 + (threadIdx.x >> 5);
  const int tile  = wave & 63;
  const int split = wave >> 6;
  const int n0    = tile * 16;
  const int m     = lane & 15;
  const int khalf = lane >> 4;

  const float* arow = xin + (size_t)m * K;            // batch row (L2-resident)
  const float* brow = wgt + (size_t)(n0 + m) * K;     // fc_w row for col n==m
  const int kbeg = split * CHUNK;

  v8f acc = {};
  for (int kk = kbeg; kk < kbeg + CHUNK; kk += 4) {
    const int k0 = kk + 2 * khalf;
    v2f a = *reinterpret_cast<const v2f*>(arow + k0);
    v2f b = __builtin_nontemporal_load(reinterpret_cast<const v2f*>(brow + k0));
    acc = wmma4(a, b, acc);
  }
#pragma unroll
  for (int r = 0; r < 8; ++r) {
    const int bat = r + 8 * khalf;
    atomicAdd(&accum[bat * 1024 + n0 + m], acc[r]);
  }
}

__global__ __launch_bounds__(256) void k_zero(float* __restrict__ p) {
  p[blockIdx.x * 256 + threadIdx.x] = 0.0f;
}

__global__ __launch_bounds__(256) void k_final(const float* __restrict__ acc,
                                               const float* __restrict__ bias,
                                               float* __restrict__ out) {
  int idx = blockIdx.x * 256 + threadIdx.x;   // 16*1024
  int f = idx & 1023;
  out[idx] = fmaxf(acc[idx] + bias[f], 0.0f);
}

extern "C" void kernel_launch(void* const* d_in, const int* in_sizes, int n_in,
                              void* d_out, int out_size, void* d_ws, size_t ws_size,
                              hipStream_t stream) {
  (void)in_sizes; (void)n_in; (void)out_size; (void)ws_size;
  const float* features = (const float*)d_in[0];
  const float* c1w = (const float*)d_in[1];
  const float* c1b = (const float*)d_in[2];
  const float* c2w = (const float*)d_in[3];
  const float* c2b = (const float*)d_in[4];
  const float* l1w = (const float*)d_in[5];
  const float* l1b = (const float*)d_in[6];
  const float* l2w = (const float*)d_in[7];
  const float* l2b = (const float*)d_in[8];
  const float* l3w = (const float*)d_in[9];
  const float* l3b = (const float*)d_in[10];
  const float* fcw = (const float*)d_in[11];
  const float* fcb = (const float*)d_in[12];
  float* out = (float*)d_out;

  // workspace carve-out (~52 MB)
  char* ws = (char*)d_ws;
  size_t off = 0;
  auto carve = [&](size_t nfloats) {
    float* p = (float*)(ws + off);
    off += ((nfloats * sizeof(float) + 255) / 256) * 256;
    return p;
  };
  float* x1  = carve((size_t)16 * 32 * 102 * 102);  // conv1 out
  float* xp  = carve((size_t)16 * 32 * 50 * 50);    // pool out
  float* x2  = carve((size_t)16 * 16 * 42 * 42);    // conv2 out
  float* x3  = carve((size_t)16 * 64 * 38 * 38);    // local1 out
  float* x4  = carve((size_t)16 * 128 * 34 * 34);   // local2 out
  float* x5  = carve((size_t)16 * 256 * 22 * 22);   // local3 out
  float* acc = carve((size_t)16 * 1024);            // FC accumulator

  // conv1: M = 16*102*102 = 166464 -> 10404 M-tiles, MT=2 -> 5202 blocks.
  // K = 363 (1 chunk), KPAD = 364, LDS = 2*16*364*4 = 46592 B.
  k_conv<11, 3, 3, 112, 112, 102, 102, 32, 2>
      <<<5202, 128, 2 * 16 * 364 * 4, stream>>>(features, c1w, c1b, x1);

  k_pool<<<5000, 256, 0, stream>>>(x1, xp);  // 1,280,000 outputs

  // conv2: M = 16*42*42 = 28224 -> 1764 M-tiles, MT=4 -> 441 blocks.
  // K = 2592, chunked by 4 channels: KCH = 324 = KPAD, 8 chunks,
  // LDS = 4*16*324*4 = 82944 B.
  k_conv<9, 32, 4, 50, 50, 42, 42, 16, 4>
      <<<441, 128, 4 * 16 * 324 * 4, stream>>>(xp, c2w, c2b, x2);

  // local1: 38x38 positions, K=400 (KPAD=404), 4 waves, LDS 25856 B
  k_local<5, 16, 42, 38, 64>
      <<<1444, 128, 16 * 404 * 4, stream>>>(x2, l1w, l1b, x3);
  // local2: 34x34 positions, K=1600 (KPAD=1604), 8 waves, LDS 102656 B
  //         (947 MB NT weight stream -> dominates total runtime)
  k_local<5, 64, 38, 34, 128>
      <<<1156, 256, 16 * 1604 * 4, stream>>>(x3, l2w, l2b, x4);
  // local3: 22x22 positions, K=1152 (KPAD=1156), 16 waves, LDS 73984 B
  k_local<3, 128, 34, 22, 256>
      <<<484, 512, 16 * 1156 * 4, stream>>>(x4, l3w, l3b, x5);

  k_zero <<<  64, 256, 0, stream>>>(acc);
  k_fc   <<<1024, 128, 0, stream>>>(x5, fcw, acc);  // 4096 split-K waves
  k_final<<<  64, 256, 0, stream>>>(acc, fcb, out);
}